// GATDecoder_4741643895113
// MI455X (gfx1250) — compile-verified
//
#include <hip/hip_runtime.h>
#include <hip/hip_bf16.h>

#define N_NODES 50000
#define N_EDGES 1600000
#define DIM     256
#define OUT     64
#define ALPHA   0.2f

typedef __attribute__((ext_vector_type(2))) float v2f;
typedef __attribute__((ext_vector_type(8))) float v8f;

// ---------------------------------------------------------------------------
// Phase 1: h = x @ W  via V_WMMA_F32_16X16X4_F32 (full fp32 precision).
// One wave handles a 16-row strip of h across all 4 column tiles (OUT=64).
// A fragment (16x4 f32, 2 VGPRs): lanes 0-15 -> K=0,1 ; lanes 16-31 -> K=2,3.
// B fragment (4x16 f32, 2 VGPRs): N = lane&15, same K split as A.
// C/D (16x16 f32, 8 VGPRs): VGPR r -> M=r (lanes 0-15) / M=r+8 (lanes 16-31).
// ---------------------------------------------------------------------------
__global__ __launch_bounds__(256) void gat_gemm_wmma(const float* __restrict__ x,
                                                     const float* __restrict__ W,
                                                     float* __restrict__ h) {
    const int wave = blockIdx.x * (blockDim.x >> 5) + (threadIdx.x >> 5);
    const int lane = threadIdx.x & 31;
    const int nstrips = N_NODES / 16;  // 3125, exact
    if (wave >= nstrips) return;       // wave-uniform: EXEC stays all-1s for WMMA

    const int m0   = wave * 16;
    const int mrow = m0 + (lane & 15);
    const int koff = (lane >> 4) << 1;   // 0 for lanes 0-15, 2 for lanes 16-31
    const int ncol = lane & 15;

    v8f acc0 = {}, acc1 = {}, acc2 = {}, acc3 = {};
    const float* __restrict__ xrow = x + (size_t)mrow * DIM;

    for (int k = 0; k < DIM; k += 4) {
        v2f a;
        a.x = xrow[k + koff];
        a.y = xrow[k + koff + 1];
        const float* __restrict__ wr0 = W + (size_t)(k + koff) * OUT;
        const float* __restrict__ wr1 = W + (size_t)(k + koff + 1) * OUT;
        v2f b0, b1, b2, b3;
        b0.x = wr0[ncol];      b0.y = wr1[ncol];
        b1.x = wr0[16 + ncol]; b1.y = wr1[16 + ncol];
        b2.x = wr0[32 + ncol]; b2.y = wr1[32 + ncol];
        b3.x = wr0[48 + ncol]; b3.y = wr1[48 + ncol];
        acc0 = __builtin_amdgcn_wmma_f32_16x16x4_f32(false, a, false, b0, (short)0, acc0, false, false);
        acc1 = __builtin_amdgcn_wmma_f32_16x16x4_f32(false, a, false, b1, (short)0, acc1, false, false);
        acc2 = __builtin_amdgcn_wmma_f32_16x16x4_f32(false, a, false, b2, (short)0, acc2, false, false);
        acc3 = __builtin_amdgcn_wmma_f32_16x16x4_f32(false, a, false, b3, (short)0, acc3, false, false);
    }

    const int rbase = m0 + ((lane >> 4) << 3);
#pragma unroll
    for (int r = 0; r < 8; ++r) {
        float* __restrict__ hrow = h + (size_t)(rbase + r) * OUT;
        hrow[ncol]      = acc0[r];
        hrow[16 + ncol] = acc1[r];
        hrow[32 + ncol] = acc2[r];
        hrow[48 + ncol] = acc3[r];
    }
}

// ---------------------------------------------------------------------------
// Phase 2: per-node scores s1[i] = h[i].a[0:64], s2[i] = h[i].a[64:128].
// One wave per node; each lane covers 2 of the 64 features; shuffle-reduce.
// ---------------------------------------------------------------------------
__global__ __launch_bounds__(256) void gat_scores(const float* __restrict__ h,
                                                  const float* __restrict__ a,
                                                  float* __restrict__ s1,
                                                  float* __restrict__ s2) {
    const int node = blockIdx.x * (blockDim.x >> 5) + (threadIdx.x >> 5);
    const int lane = threadIdx.x & 31;
    if (node >= N_NODES) return;
    const float* __restrict__ hr = h + (size_t)node * OUT;
    const float h0 = hr[lane], h1 = hr[lane + 32];
    float p1 = h0 * a[lane]       + h1 * a[lane + 32];
    float p2 = h0 * a[OUT + lane] + h1 * a[OUT + lane + 32];
#pragma unroll
    for (int off = 16; off > 0; off >>= 1) {
        p1 += __shfl_xor(p1, off, 32);
        p2 += __shfl_xor(p2, off, 32);
    }
    if (lane == 0) { s1[node] = p1; s2[node] = p2; }
}

// ---------------------------------------------------------------------------
// Zero rowsum[N] and agg[N*OUT] (deterministic per call; graph-replay safe).
// ---------------------------------------------------------------------------
__global__ void gat_zero(float* __restrict__ buf, int count) {
    int i = blockIdx.x * blockDim.x + threadIdx.x;
    if (i < count) buf[i] = 0.0f;
}

// ---------------------------------------------------------------------------
// Phase 3: edge scatter. One wave per edge.
//   logit = s1[src] + s2[dst]; e = exp(-leakyrelu(logit))
//   rowsum[src] += e;  agg[src,:] += e * h[dst,:]
// Coalesced 256B read of h[dst*64..], coalesced atomics into agg[src*64..]
// (agg+rowsum = ~13MB -> resident in 192MB L2, atomics resolve there).
// ---------------------------------------------------------------------------
__global__ __launch_bounds__(256) void gat_edges(const int* __restrict__ ei,
                                                 const float* __restrict__ h,
                                                 const float* __restrict__ s1,
                                                 const float* __restrict__ s2,
                                                 float* __restrict__ rowsum,
                                                 float* __restrict__ agg) {
    const int edge = blockIdx.x * (blockDim.x >> 5) + (threadIdx.x >> 5);
    const int lane = threadIdx.x & 31;
    if (edge >= N_EDGES) return;
    const int src = ei[edge];
    const int dst = ei[N_EDGES + edge];
    const float logit = s1[src] + s2[dst];
    const float lr = logit > 0.0f ? logit : ALPHA * logit;
    const float e = expf(-lr);
    const float* __restrict__ hj = h + (size_t)dst * OUT;
    float* __restrict__ ai = agg + (size_t)src * OUT;
    unsafeAtomicAdd(&ai[lane],      e * hj[lane]);
    unsafeAtomicAdd(&ai[lane + 32], e * hj[lane + 32]);
    if (lane == 0) unsafeAtomicAdd(&rowsum[src], e);
}

// ---------------------------------------------------------------------------
// Phase 4: out = elu(agg / rowsum)
// ---------------------------------------------------------------------------
__global__ void gat_final(const float* __restrict__ agg,
                          const float* __restrict__ rowsum,
                          float* __restrict__ out) {
    int i = blockIdx.x * blockDim.x + threadIdx.x;
    if (i >= N_NODES * OUT) return;
    const float z = agg[i] / rowsum[i >> 6];  // OUT == 64
    out[i] = z > 0.0f ? z : (expf(z) - 1.0f);
}

extern "C" void kernel_launch(void* const* d_in, const int* in_sizes, int n_in,
                              void* d_out, int out_size, void* d_ws, size_t ws_size,
                              hipStream_t stream) {
    const float* x  = (const float*)d_in[0];   // [N, DIM]
    const int*   ei = (const int*)d_in[1];     // [2, E]
    const float* W  = (const float*)d_in[2];   // [DIM, OUT]
    const float* a  = (const float*)d_in[3];   // [2*OUT]
    float* out = (float*)d_out;                // [N, OUT]

    // Workspace layout (floats): h | s1 | s2 | rowsum | agg  (~26.2 MB)
    float* h      = (float*)d_ws;
    float* s1     = h + (size_t)N_NODES * OUT;
    float* s2     = s1 + N_NODES;
    float* rowsum = s2 + N_NODES;
    float* agg    = rowsum + N_NODES;

    const int wavesPerBlock = 256 / 32;  // 8

    // 1) h = x @ W (WMMA f32)
    {
        int strips = N_NODES / 16;  // 3125
        int blocks = (strips + wavesPerBlock - 1) / wavesPerBlock;
        gat_gemm_wmma<<<blocks, 256, 0, stream>>>(x, W, h);
    }
    // 2) per-node scores
    {
        int blocks = (N_NODES + wavesPerBlock - 1) / wavesPerBlock;
        gat_scores<<<blocks, 256, 0, stream>>>(h, a, s1, s2);
    }
    // 3a) zero rowsum + agg (contiguous region: rowsum then agg)
    {
        int count = N_NODES * (OUT + 1);
        int blocks = (count + 255) / 256;
        gat_zero<<<blocks, 256, 0, stream>>>(rowsum, count);
    }
    // 3b) edge scatter
    {
        int blocks = (N_EDGES + wavesPerBlock - 1) / wavesPerBlock;
        gat_edges<<<blocks, 256, 0, stream>>>(ei, h, s1, s2, rowsum, agg);
    }
    // 4) finalize
    {
        int count = N_NODES * OUT;
        int blocks = (count + 255) / 256;
        gat_final<<<blocks, 256, 0, stream>>>(agg, rowsum, out);
    }
}